// Decoder_7739531067813
// MI455X (gfx1250) — compile-verified
//
#include <hip/hip_runtime.h>

// Problem dims (fixed by reference)
#define E_ 1024
#define H_ 1024
#define B_ 64
#define T_ 32
#define S_ 64
#define C_ 2048   // = 2*H

typedef float v2f __attribute__((ext_vector_type(2)));
typedef float v8f __attribute__((ext_vector_type(8)));

// --------------------------- CDNA5 async-to-LDS ----------------------------
#if defined(__has_builtin)
#  if __has_builtin(__builtin_amdgcn_global_load_async_to_lds_b128)
#    define HAVE_ASYNC_LDS 1
#  endif
#  if __has_builtin(__builtin_amdgcn_s_wait_asynccnt)
#    define HAVE_WAIT_ASYNC 1
#  endif
#endif

// Probe-discovered signature (round-2 diagnostic): parameters are pointers to
// GCC-vector int4 in AS1 (global) / AS3 (LDS).
typedef int v4i_vs __attribute__((vector_size(16)));
typedef __attribute__((address_space(1))) v4i_vs as1_v4i;
typedef __attribute__((address_space(3))) v4i_vs as3_v4i;

__device__ __forceinline__ void async_copy_b128(const float* g, float* lds) {
#if defined(HAVE_ASYNC_LDS)
  // generic->AS1 is bit-identical; low 32 bits of a generic LDS pointer are
  // the LDS offset (AS3 pointers are 32-bit). Integer round-trip avoids
  // addrspace-cast diagnostics.
  __builtin_amdgcn_global_load_async_to_lds_b128(
      (as1_v4i*)(uintptr_t)g,
      (as3_v4i*)(unsigned int)(uintptr_t)lds,
      /*offset=*/0, /*cpol=*/0);
#else
  *(float4*)lds = *(const float4*)g;
#endif
}

template <int N>
__device__ __forceinline__ void wait_async_le() {
#if defined(HAVE_ASYNC_LDS)
#  if defined(HAVE_WAIT_ASYNC)
  __builtin_amdgcn_s_wait_asynccnt(N);
#  else
  asm volatile("s_wait_asynccnt %0" ::"i"(N) : "memory");
#  endif
#endif
}

// Native fp32 WMMA: D(16x16,f32) = A(16x4,f32) * B(4x16,f32) + C
__device__ __forceinline__ v8f wmma4f32(v2f a, v2f b, v8f c) {
  return __builtin_amdgcn_wmma_f32_16x16x4_f32(
      /*neg_a=*/false, a, /*neg_b=*/false, b,
      /*c_mod=*/(short)0, c, /*reuse_a=*/false, /*reuse_b=*/false);
}

// Fast transcendentals on v_exp_f32 (numerically stable tanh form)
__device__ __forceinline__ float fast_tanh(float x) {
  const float t = __expf(-2.0f * fabsf(x));
  return copysignf((1.0f - t) / (1.0f + t), x);
}
__device__ __forceinline__ float fast_sigmoid(float x) {
  return 1.0f / (1.0f + __expf(-x));
}

// ---------------------------------------------------------------------------
// fp32 WMMA GEMM: D[M,N] (=|+=) A[M,K] * B[K,N] (+ bias[N]) (tanh?)
// 256 threads = 8 waves; block tile 64(M) x 128(N); each wave a 64x16 strip.
// A is shared by all 8 waves -> staged in LDS, double-buffered via
// GLOBAL_LOAD_ASYNC_TO_LDS_B128 in K-chunks of 64. B columns are per-wave
// (disjoint per block) -> direct global loads, L2-resident.
// flags: bit0 = accumulate from existing D, bit1 = tanh epilogue.
// Requires M%64==0, N%128==0, K%64==0 (true for all uses here).
// ---------------------------------------------------------------------------
#define KC_ 64           // K-chunk staged per buffer
#define APAD_ 68         // padded row stride (floats): conflict-free ds_load_b64

__global__ __launch_bounds__(256) void gemm_wmma_f32(
    const float* __restrict__ A, const float* __restrict__ Bm,
    float* __restrict__ D, const float* __restrict__ bias,
    int K, int lda, int ldb, int ldd, int flags)
{
  __shared__ float Atile[2][64 * APAD_];   // 2 x 17 KB

  const int tid  = threadIdx.x;
  const int wave = tid >> 5;
  const int lane = tid & 31;
  const int half = lane >> 4;      // which 16-lane half of the wave
  const int ml   = lane & 15;      // row (A) / col (B,D) within tile
  const int m0   = blockIdx.x * 64;
  const int n    = blockIdx.y * 128 + wave * 16 + ml;  // this lane's column

  v8f acc[4];
  if (flags & 1) {
#pragma unroll
    for (int j = 0; j < 4; ++j)
#pragma unroll
      for (int i = 0; i < 8; ++i)
        acc[j][i] = D[(size_t)(m0 + j*16 + half*8 + i) * ldd + n];
  } else {
    v8f z = {};
#pragma unroll
    for (int j = 0; j < 4; ++j) acc[j] = z;
  }

  // Stage one 64 x 64 A-chunk: 1024 16B slots, 4 per thread.
  auto issue_chunk = [&](int kc, int buf) {
#pragma unroll
    for (int i = 0; i < 4; ++i) {
      const int s   = tid + i * 256;
      const int row = s >> 4;           // 0..63
      const int col = (s & 15) * 4;     // 0..60
      async_copy_b128(A + (size_t)(m0 + row) * lda + kc * KC_ + col,
                      &Atile[buf][row * APAD_ + col]);
    }
  };

  const int nchunks = K / KC_;
  issue_chunk(0, 0);

  for (int kc = 0; kc < nchunks; ++kc) {
    const int buf = kc & 1;
    if (kc + 1 < nchunks) {
      issue_chunk(kc + 1, buf ^ 1);   // prefetch next chunk
      wait_async_le<4>();             // chunk kc done; next still in flight
    } else {
      wait_async_le<0>();
    }
    __syncthreads();                  // chunk kc visible to all waves

    const float* As = &Atile[buf][0];
#pragma unroll
    for (int kk = 0; kk < KC_; kk += 4) {
      const int ka = kk + half * 2;
      const int kg = kc * KC_ + ka;
      v2f bf;
      bf.x = Bm[(size_t)kg       * ldb + n];
      bf.y = Bm[(size_t)(kg + 1) * ldb + n];
#pragma unroll
      for (int j = 0; j < 4; ++j) {
        v2f af = *(const v2f*)(As + (j * 16 + ml) * APAD_ + ka);
        acc[j] = wmma4f32(af, bf, acc[j]);
      }
    }
    __syncthreads();                  // all waves done reading before reuse
  }

  const float bv = bias ? bias[n] : 0.0f;
#pragma unroll
  for (int j = 0; j < 4; ++j)
#pragma unroll
    for (int i = 0; i < 8; ++i) {
      float v = acc[j][i] + bv;
      if (flags & 2) v = fast_tanh(v);
      D[(size_t)(m0 + j*16 + half*8 + i) * ldd + n] = v;
    }
}

// ---------------------------------------------------------------------------
// Embedding gather: emb[row,:] = emb_table[y[row],:], row = t*B+b
// ---------------------------------------------------------------------------
__global__ void gather_emb(const int* __restrict__ y,
                           const float* __restrict__ table,
                           float* __restrict__ emb)
{
  const int row = blockIdx.x;
  const int v = y[row];
  const float4* src = (const float4*)(table + (size_t)v * E_);
  float4* dst = (float4*)(emb + (size_t)row * E_);
  for (int e = threadIdx.x; e < E_ / 4; e += blockDim.x) dst[e] = src[e];
}

// ---------------------------------------------------------------------------
// GRU gate combine: h = (1-z)*n + z*h_prev with (r,z,n) gate order
// ---------------------------------------------------------------------------
__global__ void gru_combine(const float* __restrict__ gi,
                            const float* __restrict__ gh,
                            const float* __restrict__ hprev,
                            float* __restrict__ hout)
{
  const int idx = blockIdx.x * blockDim.x + threadIdx.x;
  if (idx >= B_ * H_) return;
  const int b = idx / H_, j = idx % H_;
  const float* gib = gi + (size_t)b * 3 * H_;
  const float* ghb = gh + (size_t)b * 3 * H_;
  const float r  = fast_sigmoid(gib[j]        + ghb[j]);
  const float z  = fast_sigmoid(gib[H_ + j]   + ghb[H_ + j]);
  const float nn = fast_tanh   (gib[2*H_ + j] + r * ghb[2*H_ + j]);
  hout[idx] = (1.0f - z) * nn + z * hprev[idx];
}

// ---------------------------------------------------------------------------
// Additive attention for one step: one block per batch element b.
// ---------------------------------------------------------------------------
__global__ __launch_bounds__(256) void attention_step(
    const float* __restrict__ keycache,   // [B,S,H]
    const float* __restrict__ q,          // [B,H]
    const float* __restrict__ v_att,      // [H]
    const float* __restrict__ context,    // [B,S,C]
    const int*   __restrict__ mask,       // [B,S] 1 = pad
    float* __restrict__ attn)             // [B,C]
{
  const int b = blockIdx.x;
  __shared__ float sc[S_];
  __shared__ float pr[S_];
  const int wave = threadIdx.x >> 5;
  const int lane = threadIdx.x & 31;
  const float* qb = q + (size_t)b * H_;

  for (int s = wave; s < S_; s += 8) {
    const float* kc = keycache + ((size_t)b * S_ + s) * H_;
    float acc = 0.0f;
    for (int h = lane; h < H_; h += 32)
      acc += fast_tanh(kc[h] + qb[h]) * v_att[h];
#pragma unroll
    for (int off = 16; off > 0; off >>= 1)
      acc += __shfl_down(acc, off, 32);
    if (lane == 0)
      sc[s] = mask[b * S_ + s] ? -3.402823466e38f : acc;
  }
  __syncthreads();

  // small S=64 softmax: every thread redundantly reduces (cheap, no sync tree)
  float mx = -3.402823466e38f;
  for (int s = 0; s < S_; ++s) mx = fmaxf(mx, sc[s]);
  float sum = 0.0f;
  for (int s = 0; s < S_; ++s) sum += __expf(sc[s] - mx);
  const float inv = 1.0f / sum;
  if (threadIdx.x < S_) pr[threadIdx.x] = __expf(sc[threadIdx.x] - mx) * inv;
  __syncthreads();

  for (int c = threadIdx.x; c < C_; c += 256) {
    const float* cb = context + (size_t)b * S_ * C_ + c;
    float acc = 0.0f;
    for (int s = 0; s < S_; ++s) acc += pr[s] * cb[(size_t)s * C_];
    attn[(size_t)b * C_ + c] = acc;
  }
}

__global__ void copy_f32(const float* __restrict__ src, float* __restrict__ dst, int n) {
  const int i = blockIdx.x * blockDim.x + threadIdx.x;
  if (i < n) dst[i] = src[i];
}

// ---------------------------------------------------------------------------
extern "C" void kernel_launch(void* const* d_in, const int* in_sizes, int n_in,
                              void* d_out, int out_size, void* d_ws, size_t ws_size,
                              hipStream_t stream) {
  const int*   y         = (const int*)  d_in[0];   // [T,B]
  const float* context   = (const float*)d_in[1];   // [B,S,C]
  const int*   cmask     = (const int*)  d_in[2];   // [B,S]
  const float* hidden    = (const float*)d_in[3];   // [B,H]
  const float* emb_table = (const float*)d_in[4];   // [V,E]
  const float* Wx1   = (const float*)d_in[5];
  const float* Wh1   = (const float*)d_in[6];
  const float* b_ih1 = (const float*)d_in[7];
  const float* b_hh1 = (const float*)d_in[8];
  const float* Wq    = (const float*)d_in[9];
  const float* Wk    = (const float*)d_in[10];
  const float* v_att = (const float*)d_in[11];
  const float* Wx2   = (const float*)d_in[12];
  const float* Wh2   = (const float*)d_in[13];
  const float* b_ih2 = (const float*)d_in[14];
  const float* b_hh2 = (const float*)d_in[15];
  const float* W_in  = (const float*)d_in[16];
  const float* b_in  = (const float*)d_in[17];
  const float* W_hid = (const float*)d_in[18];
  const float* b_hid = (const float*)d_in[19];
  const float* W_ctx = (const float*)d_in[20];
  const float* b_ctx = (const float*)d_in[21];

  float* out = (float*)d_out;  // logits [T*B,E] then hidden_final [B,H]

  // workspace carve-up (floats); total ~19.5M floats ~78 MB
  float* ws = (float*)d_ws;
  size_t off = 0;
  float* emb      = ws + off; off += (size_t)T_ * B_ * E_;       // gathered embeddings
  float* keyc     = ws + off; off += (size_t)B_ * S_ * H_;       // attention key cache
  float* Gx1      = ws + off; off += (size_t)T_ * B_ * 3 * H_;   // precomputed x-gates GRU1
  float* attn_all = ws + off; off += (size_t)T_ * B_ * C_;       // attn per step
  float* out_all  = ws + off; off += (size_t)T_ * B_ * H_;       // h2 per step
  float* h1buf    = ws + off; off += (size_t)B_ * H_;
  float* qbuf     = ws + off; off += (size_t)B_ * H_;
  float* Gh1      = ws + off; off += (size_t)B_ * 3 * H_;
  float* Gx2      = ws + off; off += (size_t)B_ * 3 * H_;
  float* Gh2      = ws + off; off += (size_t)B_ * 3 * H_;
  (void)ws_size; (void)in_sizes; (void)n_in; (void)out_size;

  const dim3 blk(256);

  // ---- Phase A: hoisted, fully parallel GEMMs ----
  gather_emb<<<T_ * B_, blk, 0, stream>>>(y, emb_table, emb);

  // key_cache[b,s,:] = context[b,s,:] @ Wk   ([4096,2048] x [2048,1024])
  gemm_wmma_f32<<<dim3(B_ * S_ / 64, H_ / 128), blk, 0, stream>>>(
      context, Wk, keyc, nullptr, C_, C_, H_, H_, 0);

  // Gx1 = emb @ Wx1 + b_ih1   ([2048,1024] x [1024,3072])
  gemm_wmma_f32<<<dim3(T_ * B_ / 64, 3 * H_ / 128), blk, 0, stream>>>(
      emb, Wx1, Gx1, b_ih1, E_, E_, 3 * H_, 3 * H_, 0);

  // ---- Phase B: sequential scan over T ----
  const float* h = hidden;
  for (int t = 0; t < T_; ++t) {
    // Gh1 = h @ Wh1 + b_hh1
    gemm_wmma_f32<<<dim3(1, 3 * H_ / 128), blk, 0, stream>>>(
        h, Wh1, Gh1, b_hh1, H_, H_, 3 * H_, 3 * H_, 0);
    // h1 = GRU(Gx1[t], Gh1, h)
    gru_combine<<<(B_ * H_ + 255) / 256, blk, 0, stream>>>(
        Gx1 + (size_t)t * B_ * 3 * H_, Gh1, h, h1buf);
    // q = h1 @ Wq
    gemm_wmma_f32<<<dim3(1, H_ / 128), blk, 0, stream>>>(
        h1buf, Wq, qbuf, nullptr, H_, H_, H_, H_, 0);
    // additive attention -> attn_all[t]
    float* attn_t = attn_all + (size_t)t * B_ * C_;
    attention_step<<<B_, blk, 0, stream>>>(keyc, qbuf, v_att, context, cmask, attn_t);
    // Gx2 = attn @ Wx2 + b_ih2   (K = 2048)
    gemm_wmma_f32<<<dim3(1, 3 * H_ / 128), blk, 0, stream>>>(
        attn_t, Wx2, Gx2, b_ih2, C_, C_, 3 * H_, 3 * H_, 0);
    // Gh2 = h1 @ Wh2 + b_hh2
    gemm_wmma_f32<<<dim3(1, 3 * H_ / 128), blk, 0, stream>>>(
        h1buf, Wh2, Gh2, b_hh2, H_, H_, 3 * H_, 3 * H_, 0);
    // h2 = GRU(Gx2, Gh2, h1) -> out_all[t]; becomes next h
    float* h2 = out_all + (size_t)t * B_ * H_;
    gru_combine<<<(B_ * H_ + 255) / 256, blk, 0, stream>>>(Gx2, Gh2, h1buf, h2);
    h = h2;
  }

  // ---- Phase C: fused output projection, accumulated in d_out ----
  const dim3 gridL(T_ * B_ / 64, E_ / 128);
  // out  = emb @ W_in + b_in
  gemm_wmma_f32<<<gridL, blk, 0, stream>>>(emb, W_in, out, b_in, E_, E_, E_, E_, 0);
  // out += out_all @ W_hid + b_hid
  gemm_wmma_f32<<<gridL, blk, 0, stream>>>(out_all, W_hid, out, b_hid, H_, H_, E_, E_, 1);
  // out += attn_all @ W_ctx + b_ctx; then tanh
  gemm_wmma_f32<<<gridL, blk, 0, stream>>>(attn_all, W_ctx, out, b_ctx, C_, C_, E_, E_, 1 | 2);

  // hidden_final = h2 at t = T-1
  copy_f32<<<(B_ * H_ + 255) / 256, blk, 0, stream>>>(
      out_all + (size_t)(T_ - 1) * B_ * H_, out + (size_t)T_ * B_ * E_, B_ * H_);
}